// ExtGNNLayer_86603720557216
// MI455X (gfx1250) — compile-verified
//
#include <hip/hip_runtime.h>

#define ENT_DIM  128
#define REL_DIM  128
#define CMP_DIM  256   // REL_DIM + ENT_DIM
#define NUM_RELC 200
#define MTILES   4     // 4 x 16 = 64 edges per block

typedef __attribute__((ext_vector_type(2))) float v2f;
typedef __attribute__((ext_vector_type(8))) float v8f;

// D = A(16x4 f32) * B(4x16 f32) + C(16x16 f32), wave32, fp32 accumulate.
__device__ __forceinline__ v8f wmma4_f32(v2f a, v2f b, v8f c) {
    return __builtin_amdgcn_wmma_f32_16x16x4_f32(
        /*neg_a=*/false, a, /*neg_b=*/false, b,
        /*c_mod=*/(short)0, c, /*reuse_a=*/false, /*reuse_b=*/false);
}

// ---------------------------------------------------------------------------
// Kernel 1: per-edge message GEMM + scatter-add into agg / deg.
// Block = 256 threads = 8 waves, 64 edges (4 row-tiles of 16) per block.
// Wave w computes output columns [16w, 16w+16) for all 4 row-tiles, so the
// B fragments (W_O / W_I rows) are loaded once per K-step and reused by
// 4 row-tiles x 2 paths = 8 WMMAs. Two accumulator sets (O / I path),
// per-row select in the epilogue.
// ---------------------------------------------------------------------------
__global__ __launch_bounds__(256)
void edge_msg_kernel(const int*   __restrict__ src,
                     const int*   __restrict__ dst,
                     const int*   __restrict__ etypes,
                     const float* __restrict__ ent_emb,
                     const float* __restrict__ rel_emb,
                     const float* __restrict__ Wo,
                     const float* __restrict__ bo,
                     const float* __restrict__ Wi,
                     const float* __restrict__ bi,
                     float*       __restrict__ agg,
                     float*       __restrict__ deg,
                     int E)
{
    __shared__ int s_et [MTILES * 16];
    __shared__ int s_dst[MTILES * 16];

    const int  tid   = threadIdx.x;
    const int  lane  = tid & 31;
    const int  wave  = tid >> 5;
    const int  m     = lane & 15;          // A-fragment row within a tile
    const int  khalf = (lane >> 4) << 1;   // 0 or 2: which K pair this lane holds
    const long base  = (long)blockIdx.x * (MTILES * 16);

    // stage edge metadata for the 64 edges of this block
    if (tid < MTILES * 16) {
        long e  = base + tid;
        int  ec = (e < E) ? (int)e : (E - 1);
        s_et [tid] = etypes[ec];
        s_dst[tid] = dst[ec];
    }
    __syncthreads();

    // per-lane A row pointers for each of the 4 row-tiles
    const float* arel[MTILES];
    const float* aent[MTILES];
    #pragma unroll
    for (int t = 0; t < MTILES; ++t) {
        long e  = base + t * 16 + m;
        int  ec = (e < E) ? (int)e : (E - 1);
        arel[t] = rel_emb + (long)etypes[ec] * REL_DIM;
        aent[t] = ent_emb + (long)src[ec]    * ENT_DIM;
    }

    const int    n      = (wave << 4) + m;           // output column 0..127
    const float* wo_row = Wo + (long)n * CMP_DIM;    // W[n][k] contiguous in k
    const float* wi_row = Wi + (long)n * CMP_DIM;

    v8f acc_o[MTILES], acc_i[MTILES];
    #pragma unroll
    for (int t = 0; t < MTILES; ++t) {
        acc_o[t] = (v8f){0.f,0.f,0.f,0.f,0.f,0.f,0.f,0.f};
        acc_i[t] = (v8f){0.f,0.f,0.f,0.f,0.f,0.f,0.f,0.f};
    }

    // K = 0..127 : relation-embedding half of comp_h
    #pragma unroll 2
    for (int k = 0; k < REL_DIM; k += 4) {
        v2f bO = *(const v2f*)(wo_row + k + khalf);
        v2f bI = *(const v2f*)(wi_row + k + khalf);
        v2f a[MTILES];
        #pragma unroll
        for (int t = 0; t < MTILES; ++t)
            a[t] = *(const v2f*)(arel[t] + k + khalf);
        #pragma unroll
        for (int t = 0; t < MTILES; ++t) {
            acc_o[t] = wmma4_f32(a[t], bO, acc_o[t]);
            acc_i[t] = wmma4_f32(a[t], bI, acc_i[t]);
        }
    }
    // K = 128..255 : source-entity half of comp_h
    #pragma unroll 2
    for (int k = 0; k < ENT_DIM; k += 4) {
        v2f bO = *(const v2f*)(wo_row + REL_DIM + k + khalf);
        v2f bI = *(const v2f*)(wi_row + REL_DIM + k + khalf);
        v2f a[MTILES];
        #pragma unroll
        for (int t = 0; t < MTILES; ++t)
            a[t] = *(const v2f*)(aent[t] + k + khalf);
        #pragma unroll
        for (int t = 0; t < MTILES; ++t) {
            acc_o[t] = wmma4_f32(a[t], bO, acc_o[t]);
            acc_i[t] = wmma4_f32(a[t], bI, acc_i[t]);
        }
    }

    // Epilogue: per-row O/I select, bias, scatter-add.
    // C/D layout: VGPR j, lanes 0-15 -> row j; lanes 16-31 -> row j+8; col = lane&15.
    const float bob   = bo[n];
    const float bib   = bi[n];
    const int   rhalf = (lane >> 4) << 3;   // 0 or 8

    #pragma unroll
    for (int t = 0; t < MTILES; ++t) {
        #pragma unroll
        for (int j = 0; j < 8; ++j) {
            const int  me = t * 16 + j + rhalf;
            const long ee = base + me;
            if (ee < E) {
                const bool  is_o = s_et[me] < NUM_RELC;
                const float v    = is_o ? (acc_o[t][j] + bob) : (acc_i[t][j] + bib);
                atomicAdd(&agg[(long)s_dst[me] * ENT_DIM + n], v);
            }
        }
    }

    // one degree increment per edge (threads 0..63 of the block)
    if (tid < MTILES * 16 && (base + tid) < E)
        atomicAdd(&deg[s_dst[tid]], 1.0f);
}

// ---------------------------------------------------------------------------
// Kernel 2/3: out = A @ W^T + b  (+ agg/max(deg,1) when agg != nullptr).
// Block = 256 threads = 8 waves; tile = 16 rows x 128 cols; K = 128.
// Used for ent_new (with agg/deg) and rel_new (agg == nullptr).
// ---------------------------------------------------------------------------
__global__ __launch_bounds__(256)
void gemm_bias_kernel(const float* __restrict__ A,
                      const float* __restrict__ W,
                      const float* __restrict__ b,
                      const float* __restrict__ agg,
                      const float* __restrict__ deg,
                      float*       __restrict__ out,
                      int N)
{
    const int  tid   = threadIdx.x;
    const int  lane  = tid & 31;
    const int  wave  = tid >> 5;
    const int  m     = lane & 15;
    const int  khalf = (lane >> 4) << 1;
    const long tile  = (long)blockIdx.x * 16;

    const long r    = tile + m;
    const long rc   = (r < N) ? r : (long)(N - 1);
    const float* ar = A + rc * ENT_DIM;

    const int    n  = (wave << 4) + m;
    const float* wr = W + (long)n * ENT_DIM;

    v8f acc = {0.f,0.f,0.f,0.f,0.f,0.f,0.f,0.f};

    #pragma unroll 4
    for (int k = 0; k < ENT_DIM; k += 4) {
        v2f av = *(const v2f*)(ar + k + khalf);
        v2f bv = *(const v2f*)(wr + k + khalf);
        acc = wmma4_f32(av, bv, acc);
    }

    const float bn    = b[n];
    const int   rhalf = (lane >> 4) << 3;

    #pragma unroll
    for (int j = 0; j < 8; ++j) {
        const int  me = j + rhalf;
        const long rr = tile + me;
        if (rr < N) {
            float v = acc[j] + bn;
            if (agg != nullptr) {
                float d = deg[rr];
                d = (d > 1.0f) ? d : 1.0f;
                v += agg[rr * ENT_DIM + n] / d;
            }
            out[rr * ENT_DIM + n] = v;
        }
    }
}

__global__ void zero_kernel(float* __restrict__ p, long n)
{
    const long i = (long)blockIdx.x * blockDim.x + threadIdx.x;
    if (i < n) p[i] = 0.0f;
}

// ---------------------------------------------------------------------------
extern "C" void kernel_launch(void* const* d_in, const int* in_sizes, int n_in,
                              void* d_out, int out_size, void* d_ws, size_t ws_size,
                              hipStream_t stream)
{
    const int*   src     = (const int*)  d_in[0];
    const int*   dst     = (const int*)  d_in[1];
    const int*   etypes  = (const int*)  d_in[2];
    const float* ent_emb = (const float*)d_in[3];
    const float* rel_emb = (const float*)d_in[4];
    const float* Wo      = (const float*)d_in[5];
    const float* bo      = (const float*)d_in[6];
    const float* Wi      = (const float*)d_in[7];
    const float* bi      = (const float*)d_in[8];
    const float* Ws      = (const float*)d_in[9];
    const float* bs      = (const float*)d_in[10];
    const float* Wr      = (const float*)d_in[11];
    const float* br      = (const float*)d_in[12];

    const int E = in_sizes[0];             // 800000
    const int N = in_sizes[3] / ENT_DIM;   // 50000
    const int R = in_sizes[4] / REL_DIM;   // 400 (= 2*NUM_REL)

    // workspace: agg_sum [N,128] + deg [N]
    float* agg = (float*)d_ws;
    float* deg = agg + (size_t)N * ENT_DIM;

    const long zn = (long)N * ENT_DIM + N;
    zero_kernel<<<(int)((zn + 255) / 256), 256, 0, stream>>>(agg, zn);

    const int edges_per_block = MTILES * 16;
    edge_msg_kernel<<<(E + edges_per_block - 1) / edges_per_block, 256, 0, stream>>>(
        src, dst, etypes, ent_emb, rel_emb, Wo, bo, Wi, bi, agg, deg, E);

    float* out_ent = (float*)d_out;
    float* out_rel = out_ent + (size_t)N * ENT_DIM;

    gemm_bias_kernel<<<(N + 15) / 16, 256, 0, stream>>>(
        ent_emb, Ws, bs, agg, deg, out_ent, N);

    gemm_bias_kernel<<<(R + 15) / 16, 256, 0, stream>>>(
        rel_emb, Wr, br, nullptr, nullptr, out_rel, R);
}